// ODMLoss_26714696581696
// MI455X (gfx1250) — compile-verified
//
#include <hip/hip_runtime.h>
#include <hip/hip_bf16.h>
#include <math.h>

#define BB   64
#define PP   16320
#define NOBJ 32
#define NCLS 21
#define CE_TILE 256

typedef __attribute__((ext_vector_type(2))) float v2f;
typedef __attribute__((ext_vector_type(8))) float v8f;

// ---------- helpers ----------

// order-preserving map float -> uint (monotone for all floats)
__device__ __forceinline__ unsigned ordf(float f) {
    unsigned b = __float_as_uint(f);
    return (b & 0x80000000u) ? ~b : (b | 0x80000000u);
}
__device__ __forceinline__ float unordf(unsigned u) {
    unsigned b = (u & 0x80000000u) ? (u & 0x7FFFFFFFu) : ~u;
    return __uint_as_float(b);
}

// Sum 64 fp32 values held as a.x/a.y across a full wave using one
// V_WMMA_F32_16X16X4_F32 with B = ones (row-sums of A land in D).
// Requires all 32 lanes active. Every lane returns the total.
__device__ __forceinline__ float wave_sum64_wmma(v2f a) {
    v2f ones; ones.x = 1.0f; ones.y = 1.0f;
    v8f c = {};
    c = __builtin_amdgcn_wmma_f32_16x16x4_f32(false, a, false, ones,
                                              (short)0, c, false, false);
    float s = c[0] + c[1] + c[2] + c[3] + c[4] + c[5] + c[6] + c[7];
    s += __shfl_xor(s, 16, 32);   // lanes 0-15 hold rows 0-7, 16-31 rows 8-15
    return s;
}

// Block (256 threads) sum via 4 chained WMMA accumulations in wave 0.
// Result valid on threads 0..31 (lane-uniform); others return 0.
__device__ __forceinline__ float block_sum_256(float v, float* red) {
    red[threadIdx.x] = v;
    __syncthreads();
    float total = 0.0f;
    if (threadIdx.x < 32) {
        v2f ones; ones.x = 1.0f; ones.y = 1.0f;
        v8f c = {};
#pragma unroll
        for (int i = 0; i < 4; ++i) {
            v2f a;
            a.x = red[threadIdx.x + i * 64];
            a.y = red[threadIdx.x + 32 + i * 64];
            c = __builtin_amdgcn_wmma_f32_16x16x4_f32(false, a, false, ones,
                                                      (short)0, c, false, false);
        }
        float s = c[0] + c[1] + c[2] + c[3] + c[4] + c[5] + c[6] + c[7];
        s += __shfl_xor(s, 16, 32);
        total = s;
    }
    __syncthreads();
    return total;
}

struct Dec { float cx, cy, w, h; };
__device__ __forceinline__ Dec decode_arm(float4 l, float4 pr) {
    Dec d;
    d.cx = pr.x + l.x * 0.1f * pr.z;
    d.cy = pr.y + l.y * 0.1f * pr.w;
    d.w  = pr.z * __expf(l.z * 0.2f);
    d.h  = pr.w * __expf(l.w * 0.2f);
    return d;
}

// ---------- kernels ----------

__global__ void k_init(int* num_pos, float* accum) {
    int t = threadIdx.x;
    if (t < BB) num_pos[t] = 0;
    if (t < 2)  accum[t] = 0.0f;
}

// One block per image: per-prior best truth (IoU, reserve-masked) and
// per-truth best prior (packed u64 LDS atomicMax, first-index ties).
__global__ void k_match(const float* __restrict__ arm_loc,
                        const float* __restrict__ arm_conf,
                        const float* __restrict__ priors,
                        const float* __restrict__ truths,
                        float* __restrict__ bt_ov,
                        int*   __restrict__ bt_idx,
                        int*   __restrict__ bp_idx) {
    const int b = blockIdx.x;
    __shared__ float4 s_tr[NOBJ];
    __shared__ unsigned long long s_best[NOBJ];
    const int tid = threadIdx.x;
    if (tid < NOBJ) {
        s_tr[tid]   = ((const float4*)truths)[b * NOBJ + tid];
        s_best[tid] = 0ull;
    }
    __syncthreads();

    for (int p = tid; p < PP; p += blockDim.x) {
        const size_t bp = (size_t)b * PP + p;
        float2 cc = ((const float2*)arm_conf)[bp];
        bool reserve = (1.0f / (1.0f + __expf(cc.x - cc.y))) > 0.01f;
        float4 l  = ((const float4*)arm_loc)[bp];
        float4 pr = ((const float4*)priors)[p];
        Dec d = decode_arm(l, pr);
        float ax1 = d.cx - 0.5f * d.w, ay1 = d.cy - 0.5f * d.h;
        float ax2 = d.cx + 0.5f * d.w, ay2 = d.cy + 0.5f * d.h;
        float areaB = d.w * d.h;

        float bestv = -2.0f; int besti = 0;
#pragma unroll 4
        for (int n = 0; n < NOBJ; ++n) {
            float4 t = s_tr[n];
            float lx = fmaxf(t.x, ax1), ly = fmaxf(t.y, ay1);
            float rx = fminf(t.z, ax2), ry = fminf(t.w, ay2);
            float iw = fmaxf(rx - lx, 0.0f), ih = fmaxf(ry - ly, 0.0f);
            float inter = iw * ih;
            float areaA = (t.z - t.x) * (t.w - t.y);
            float iou = inter / (areaA + areaB - inter);
            float ov = reserve ? iou : -1.0f;
            if (ov > bestv) { bestv = ov; besti = n; }
            // pack (ordered value, PP-1-p): max -> best value, smallest p on tie
            unsigned long long key =
                ((unsigned long long)ordf(ov) << 32) | (unsigned)(PP - 1 - p);
            if (key > s_best[n]) atomicMax(&s_best[n], key);
        }
        bt_ov[bp]  = bestv;
        bt_idx[bp] = besti;
    }
    __syncthreads();
    if (tid < NOBJ)
        bp_idx[b * NOBJ + tid] = PP - 1 - (int)(s_best[tid] & 0xFFFFFFFFu);
}

// Sequential per-image scatter override (matches .at[].set last-wins order).
__global__ void k_override(const int* __restrict__ bp_idx,
                           float* __restrict__ bt_ov,
                           int*   __restrict__ bt_idx) {
    int b = threadIdx.x;
    if (b >= BB) return;
    for (int n = 0; n < NOBJ; ++n) {
        int p = bp_idx[b * NOBJ + n];
        size_t bp = (size_t)b * PP + p;
        bt_ov[bp]  = 2.0f;
        bt_idx[bp] = n;
    }
}

// conf targets, positive count, and fused smooth-L1 loc loss.
__global__ void k_targets(const float* __restrict__ arm_loc,
                          const float* __restrict__ arm_conf,
                          const float* __restrict__ odm_loc,
                          const float* __restrict__ priors,
                          const float* __restrict__ truths,
                          const int*   __restrict__ labels,
                          const float* __restrict__ bt_ov,
                          const int*   __restrict__ bt_idx,
                          int*   __restrict__ conf_t,
                          int*   __restrict__ num_pos,
                          float* __restrict__ accum) {
    const int b = blockIdx.x;
    __shared__ float4 s_tr[NOBJ];
    __shared__ int    s_lb[NOBJ];
    __shared__ float  red[256];
    const int tid = threadIdx.x;
    if (tid < NOBJ) {
        s_tr[tid] = ((const float4*)truths)[b * NOBJ + tid];
        s_lb[tid] = labels[b * NOBJ + tid];
    }
    __syncthreads();

    float sl1_acc = 0.0f, pos_acc = 0.0f;
    for (int p = tid; p < PP; p += blockDim.x) {
        const size_t bp = (size_t)b * PP + p;
        float2 cc = ((const float2*)arm_conf)[bp];
        bool reserve = (1.0f / (1.0f + __expf(cc.x - cc.y))) > 0.01f;
        float ov = bt_ov[bp];
        int   ti = bt_idx[bp];
        int conf = s_lb[ti] + 1;
        if (ov < 0.5f) conf = 0;
        if (!reserve)  conf = -1;
        conf_t[bp] = conf;
        if (conf > 0) {
            pos_acc += 1.0f;
            float4 l  = ((const float4*)arm_loc)[bp];
            float4 pr = ((const float4*)priors)[p];
            Dec d = decode_arm(l, pr);             // ref_center == decoded cx,cy,w,h
            float4 t = s_tr[ti];
            float gcx = ((t.x + t.z) * 0.5f - d.cx) / (0.1f * d.w);
            float gcy = ((t.y + t.w) * 0.5f - d.cy) / (0.1f * d.h);
            float gw  = __logf((t.z - t.x) / d.w) * 5.0f;   // / 0.2
            float gh  = __logf((t.w - t.y) / d.h) * 5.0f;
            float4 ol = ((const float4*)odm_loc)[bp];
            float dd;
            dd = fabsf(ol.x - gcx); sl1_acc += (dd < 1.0f) ? 0.5f * dd * dd : dd - 0.5f;
            dd = fabsf(ol.y - gcy); sl1_acc += (dd < 1.0f) ? 0.5f * dd * dd : dd - 0.5f;
            dd = fabsf(ol.z - gw);  sl1_acc += (dd < 1.0f) ? 0.5f * dd * dd : dd - 0.5f;
            dd = fabsf(ol.w - gh);  sl1_acc += (dd < 1.0f) ? 0.5f * dd * dd : dd - 0.5f;
        }
    }
    float tot_sl1 = block_sum_256(sl1_acc, red);
    float tot_pos = block_sum_256(pos_acc, red);
    if (tid == 0) {
        atomicAdd(&accum[0], tot_sl1);
        num_pos[b] = (int)(tot_pos + 0.5f);       // one block per image
    }
}

// Per-prior cross entropy: logsumexp - target logit.  Streams odm_conf (88 MB,
// the bandwidth-dominant pass) through LDS using CDNA5 async b128 copies:
// fully coalesced 16B transfers, no VGPR staging.  Tile = 256 rows * 84 B
// = 21504 B (always float4-exact since 4 rows = 336 B = 21 chunks).
__global__ void k_ce(const float* __restrict__ odm_conf,
                     const int*   __restrict__ conf_t,
                     float* __restrict__ ce) {
    __shared__ float s_buf[CE_TILE * NCLS];       // 21504 bytes
    const int b   = blockIdx.y;
    const int p0  = blockIdx.x * CE_TILE;
    const int valid = (PP - p0 < CE_TILE) ? (PP - p0) : CE_TILE;
    const int tid = threadIdx.x;
    const int nchunk = (valid * NCLS) / 4;        // float4 chunks, exact

    const unsigned long long gbase =
        (unsigned long long)(uintptr_t)(odm_conf + ((size_t)b * PP + p0) * NCLS);
    const unsigned lbase = (unsigned)(uintptr_t)(&s_buf[0]);

    for (int i = tid; i < nchunk; i += blockDim.x) {
        unsigned long long ga = gbase + (unsigned long long)(i * 16);
        unsigned la = lbase + (unsigned)(i * 16);
        asm volatile("global_load_async_to_lds_b128 %0, %1, off"
                     :: "v"(la), "v"(ga) : "memory");
    }
    asm volatile("s_wait_asynccnt 0x0" ::: "memory");
    __syncthreads();

    if (tid < valid) {
        const float* row = &s_buf[tid * NCLS];    // stride 21 dwords: no bank conflicts
        float x[NCLS];
        float m = -3.4e38f;
#pragma unroll
        for (int c = 0; c < NCLS; ++c) { x[c] = row[c]; m = fmaxf(m, x[c]); }
        float s = 0.0f;
#pragma unroll
        for (int c = 0; c < NCLS; ++c) s += __expf(x[c] - m);
        float lse = m + __logf(s);
        const size_t bp = (size_t)b * PP + p0 + tid;
        int ct = conf_t[bp];
        int c0 = (ct == -1) ? 0 : ct;
        c0 = (c0 < 0) ? 0 : ((c0 > NCLS - 1) ? NCLS - 1 : c0);
        ce[bp] = lse - x[c0];
    }
}

// Per-image hard-negative mining: exact k-th-largest via 4x8-bit radix select
// over the "mine" keys (0 for pos/ignore, CE for negatives), then
// loss_c_b = sum(ce | pos) + sum(ce | mine > kth) + remaining * kth.
__global__ void k_select(const int*   __restrict__ conf_t,
                         const float* __restrict__ ce_arr,
                         const int*   __restrict__ num_pos,
                         float* __restrict__ accum) {
    const int b = blockIdx.x;
    const int tid = threadIdx.x;
    __shared__ unsigned int hist[256];
    __shared__ float red[256];
    __shared__ int sh_sel, sh_rem;
    const size_t base = (size_t)b * PP;

    float posce = 0.0f;
    for (int p = tid; p < PP; p += 256) {
        int ct = conf_t[base + p];
        if (ct > 0) posce += ce_arr[base + p];
    }
    float sum_pos = block_sum_256(posce, red);

    const int np = num_pos[b];
    int k = 3 * np; if (k > PP - 1) k = PP - 1;
    unsigned prefix = 0, prefmask = 0;
    int rem0 = k;                       // tracked by tid 0

    if (k > 0) {
        for (int pass = 0; pass < 4; ++pass) {
            const int shift = 24 - 8 * pass;
            hist[tid] = 0;
            __syncthreads();
            for (int p = tid; p < PP; p += 256) {
                int ct = conf_t[base + p];
                unsigned u = (ct > 0 || ct == -1) ? 0x80000000u
                                                  : ordf(ce_arr[base + p]);
                if ((u & prefmask) == prefix)
                    atomicAdd(&hist[(u >> shift) & 0xFFu], 1u);
            }
            __syncthreads();
            if (tid == 0) {
                int cum = 0, sel = 0;
                for (int d = 255; d >= 0; --d) {
                    int h = (int)hist[d];
                    if (cum + h >= rem0) { sel = d; break; }
                    cum += h;
                }
                rem0 -= cum;
                sh_sel = sel;
                sh_rem = rem0;
            }
            __syncthreads();
            prefix   |= ((unsigned)sh_sel) << shift;
            prefmask |= 0xFFu << shift;
        }
    }

    float gtce = 0.0f;
    if (k > 0) {
        for (int p = tid; p < PP; p += 256) {
            int ct = conf_t[base + p];
            if (ct > 0 || ct == -1) continue;
            float cv = ce_arr[base + p];
            if (ordf(cv) > prefix) gtce += cv;
        }
    }
    float sum_gt = block_sum_256(gtce, red);

    if (tid == 0) {
        float lc = sum_pos + sum_gt;
        if (k > 0) lc += (float)sh_rem * unordf(prefix);
        atomicAdd(&accum[1], lc);
    }
}

// Single wave: total_num = sum(num_pos[0..63]) via one WMMA, then divide.
__global__ void k_final(const int* __restrict__ num_pos,
                        const float* __restrict__ accum,
                        float* __restrict__ out) {
    const int lane = threadIdx.x;                 // 32 threads, full wave
    v2f a;
    a.x = (float)num_pos[lane];
    a.y = (float)num_pos[lane + 32];
    float total = wave_sum64_wmma(a);
    if (lane == 0) {
        out[0] = accum[0] / total;
        out[1] = accum[1] / total;
    }
}

// ---------- launch ----------

extern "C" void kernel_launch(void* const* d_in, const int* in_sizes, int n_in,
                              void* d_out, int out_size, void* d_ws, size_t ws_size,
                              hipStream_t stream) {
    (void)in_sizes; (void)n_in; (void)out_size; (void)ws_size;
    const float* arm_loc  = (const float*)d_in[0];
    const float* arm_conf = (const float*)d_in[1];
    const float* odm_loc  = (const float*)d_in[2];
    const float* odm_conf = (const float*)d_in[3];
    const float* priors   = (const float*)d_in[4];
    const float* truths   = (const float*)d_in[5];
    const int*   labels   = (const int*)d_in[6];
    float* out = (float*)d_out;

    char* ws = (char*)d_ws;
    size_t off = 0;
    auto alloc = [&](size_t bytes) -> void* {
        void* p = ws + off;
        off += (bytes + 255) & ~(size_t)255;
        return p;
    };
    float* bt_ov   = (float*)alloc(sizeof(float) * BB * PP);
    int*   bt_idx  = (int*)  alloc(sizeof(int)   * BB * PP);
    int*   conf_t  = (int*)  alloc(sizeof(int)   * BB * PP);
    float* ce      = (float*)alloc(sizeof(float) * BB * PP);
    int*   bp_idx  = (int*)  alloc(sizeof(int)   * BB * NOBJ);
    int*   num_pos = (int*)  alloc(sizeof(int)   * BB);
    float* accum   = (float*)alloc(sizeof(float) * 2);

    k_init<<<1, 128, 0, stream>>>(num_pos, accum);
    k_match<<<BB, 256, 0, stream>>>(arm_loc, arm_conf, priors, truths,
                                    bt_ov, bt_idx, bp_idx);
    k_override<<<1, 64, 0, stream>>>(bp_idx, bt_ov, bt_idx);
    k_targets<<<BB, 256, 0, stream>>>(arm_loc, arm_conf, odm_loc, priors, truths,
                                      labels, bt_ov, bt_idx, conf_t, num_pos, accum);
    dim3 gce((PP + CE_TILE - 1) / CE_TILE, BB);
    k_ce<<<gce, CE_TILE, 0, stream>>>(odm_conf, conf_t, ce);
    k_select<<<BB, 256, 0, stream>>>(conf_t, ce, num_pos, accum);
    k_final<<<1, 32, 0, stream>>>(num_pos, accum, out);
}